// LinearSSM_77335181131929
// MI455X (gfx1250) — compile-verified
//
#include <hip/hip_runtime.h>
#include <cstdint>
#include <cstddef>

// ---------------------------------------------------------------------------
// Linear SSM on MI455X (gfx1250), fp32 WMMA chunked-scan implementation.
//   h_t = h_{t-1} @ Abar^T + x_t @ Bbar^T ;  y_t = h_t @ C^T + x_t @ D^T
// Chunked scan: pass A (parallel local scans) -> pass B (P=Abar^T^L chunk
// chain, 1 WG) -> pass C (parallel corrected scans, stream hs) -> out GEMM.
// All matrix math uses V_WMMA_F32_16X16X4_F32 (fp32 precision for the
// 4096-step recurrence).
// ---------------------------------------------------------------------------

#define BATCH 32
#define SEQT  4096
#define IND   64
#define SD    256
#define OD    64
#define CHUNK 128
#define NCHUNK (SEQT / CHUNK)   // 32

typedef float v2f __attribute__((ext_vector_type(2)));
typedef float v8f __attribute__((ext_vector_type(8)));

// workspace layout (float offsets)
#define WS_ABSW   0                        // Abar^T swizzled B-op layout [64][2][256][2]
#define WS_ABT    (WS_ABSW + 65536)        // Abar^T plain [256][256]
#define WS_BSW    (WS_ABT + 65536)         // Bbar^T swizzled [16][2][256][2]
#define WS_CSW    (WS_BSW + 16384)         // C^T swizzled [64][2][64][2]
#define WS_DSW    (WS_CSW + 16384)         // D^T swizzled [16][2][64][2]
#define WS_SCH    (WS_DSW + 4096)          // S_c local chunk-final states [NC][32][256]
#define WS_HST    (WS_SCH + 262144)        // chunk start states [NC][32][256]
#define WS_HS     (WS_HST + 262144)        // hs [32][4096][256]  (134 MB)

#define LDS_BYTES 294912                   // 256KB Abar^T + 32KB h double buffer

// ---------------------------------------------------------------------------
// K1: build dt-discretized, operand-swizzled matrices.
// Swizzle: element (kk, half, n, j) = Bmat[4*kk + 2*half + j][n] so a lane's
// two B-operand VGPRs are one contiguous 8-byte LDS/global load.
// ---------------------------------------------------------------------------
__global__ void ssm_setup(const float* __restrict__ A, const float* __restrict__ Bm,
                          const float* __restrict__ Cm, const float* __restrict__ Dm,
                          const float* __restrict__ log_dt, float* __restrict__ ws)
{
  int idx = blockIdx.x * blockDim.x + threadIdx.x;
  float dt = expf(log_dt[0]);
  dt = dt > 1.0f ? 1.0f : dt;
  if (idx < 65536) {                       // Abar^T swizzled: Bmat[k][n] = (n==k) + A[n][k]*dt
    int j = idx & 1, n = (idx >> 1) & 255, half = (idx >> 9) & 1, kk = idx >> 10;
    int k = 4 * kk + 2 * half + j;
    ws[WS_ABSW + idx] = (n == k ? 1.0f : 0.0f) + A[n * SD + k] * dt;
  } else if (idx < 131072) {               // Abar^T plain row-major [s][n]
    int i2 = idx - 65536; int s = i2 >> 8, n = i2 & 255;
    ws[WS_ABT + i2] = (s == n ? 1.0f : 0.0f) + A[n * SD + s] * dt;
  } else if (idx < 147456) {               // Bbar^T swizzled: Bmat[i][s] = B[s][i]*dt
    int i2 = idx - 131072;
    int j = i2 & 1, n = (i2 >> 1) & 255, half = (i2 >> 9) & 1, kk = i2 >> 10;
    int i = 4 * kk + 2 * half + j;
    ws[WS_BSW + i2] = Bm[n * IND + i] * dt;
  } else if (idx < 163840) {               // C^T swizzled: Bmat[s][o] = C[o][s]
    int i2 = idx - 147456;
    int j = i2 & 1, o = (i2 >> 1) & 63, half = (i2 >> 7) & 1, kk = i2 >> 8;
    int s = 4 * kk + 2 * half + j;
    ws[WS_CSW + i2] = Cm[o * SD + s];
  } else if (idx < 167936) {               // D^T swizzled: Bmat[i][o] = D[o][i]
    int i2 = idx - 163840;
    int j = i2 & 1, o = (i2 >> 1) & 63, half = (i2 >> 7) & 1, kk = i2 >> 8;
    int i = 4 * kk + 2 * half + j;
    ws[WS_DSW + i2] = Dm[o * IND + i];
  }
}

// ---------------------------------------------------------------------------
// K2/K4: within-chunk scan. One WG per (chunk, batch-group of 16).
// 16 waves; wave w owns state columns [16w, 16w+16). Per step:
//  - 16 WMMAs accumulate Bu tile (K=64, x A-ops from global, Bbar in regs)
//  - 64 WMMAs accumulate h_{t-1} @ Abar^T (A-ops from LDS h, B-ops from LDS)
//  - write h_t tile into the other LDS buffer, one barrier per step.
// mode 0: h0 = 0, emit chunk-final local state S_c.
// mode 1: h0 = Hstart[c], stream every h_t to hs.
// ---------------------------------------------------------------------------
__global__ __launch_bounds__(512) void ssm_scan(
    const float* __restrict__ x, const float* __restrict__ ws,
    float* __restrict__ Sout, const float* __restrict__ Hstart,
    float* __restrict__ hs, int mode)
{
  extern __shared__ float lds[];
  float* AbLDS = lds;            // 65536 floats: Abar^T swizzled
  float* hbuf  = lds + 65536;    // 2 x 16x256 double buffer
  const float* AbSW = ws + WS_ABSW;
  const float* BSW  = ws + WS_BSW;
  const int tid   = threadIdx.x;
  const int wave  = tid >> 5, lane = tid & 31;
  const int lhalf = lane >> 4, lm = lane & 15;
  const int chunk = blockIdx.x >> 1, bg = blockIdx.x & 1;

  for (int i = tid; i < 65536 / 4; i += 512)
    ((float4*)AbLDS)[i] = ((const float4*)AbSW)[i];
  for (int i = tid; i < 4096; i += 512)
    hbuf[i] = mode ? Hstart[(chunk * 32 + bg * 16) * SD + i] : 0.0f;
  __syncthreads();

  // resident Bbar^T slice for this wave's 16 state columns (32 VGPRs)
  v2f breg[16];
#pragma unroll
  for (int kk = 0; kk < 16; kk++)
    breg[kk] = *(const v2f*)(BSW + ((kk * 2 + lhalf) * SD + wave * 16 + lm) * 2);

  const int t0 = chunk * CHUNK;
  int p = 0;
  for (int t = 0; t < CHUNK; t++) {
    const int tt = t0 + t;
    v8f acc = {};
    // Bu_t tile: x_t (16x64) @ Bbar^T slice (64x16)
    const float* xp = x + ((size_t)(bg * 16 + lm) * SEQT + tt) * IND + lhalf * 2;
#pragma unroll
    for (int kk = 0; kk < 16; kk++) {
      v2f a = *(const v2f*)(xp + kk * 4);
      acc = __builtin_amdgcn_wmma_f32_16x16x4_f32(false, a, false, breg[kk],
                                                  (short)0, acc, false, false);
    }
    // recurrence: h_{t-1} (16x256) @ Abar^T slice (256x16)
    const float* hp = hbuf + p * 4096;
#pragma unroll
    for (int kk = 0; kk < 64; kk++) {
      v2f a = *(const v2f*)(hp + lm * SD + kk * 4 + lhalf * 2);
      v2f b = *(const v2f*)(AbLDS + ((kk * 2 + lhalf) * SD + wave * 16 + lm) * 2);
      acc = __builtin_amdgcn_wmma_f32_16x16x4_f32(false, a, false, b,
                                                  (short)0, acc, false, false);
    }
    // C/D layout: VGPR r, lanes<16 -> M=r, lanes>=16 -> M=r+8
    float* hq = hbuf + (1 - p) * 4096;
#pragma unroll
    for (int r = 0; r < 8; r++)
      hq[(r + lhalf * 8) * SD + wave * 16 + lm] = acc[r];
    if (mode) {
#pragma unroll
      for (int r = 0; r < 8; r++)
        hs[(((size_t)(bg * 16 + r + lhalf * 8)) * SEQT + tt) * SD + wave * 16 + lm] = acc[r];
    }
    __syncthreads();
    p ^= 1;
  }
  if (!mode) {
    float* sp = Sout + (chunk * 32 + bg * 16) * SD;
    for (int i = tid; i < 4096; i += 512) sp[i] = hbuf[p * 4096 + i];
  }
}

// ---------------------------------------------------------------------------
// K3: single workgroup. P = (Abar^T)^CHUNK by 7 in-LDS squarings, then the
// 32-step chunk chain h_start_{c+1} = h_start_c @ P + S_c. Emits Hstart[] and
// hidden_final.
// ---------------------------------------------------------------------------
__global__ __launch_bounds__(512) void ssm_combine(
    const float* __restrict__ ws, const float* __restrict__ hidden,
    float* __restrict__ hstart, const float* __restrict__ sch,
    float* __restrict__ hidden_final)
{
  extern __shared__ float lds[];
  float* M  = lds;            // 65536: current matrix power (row-major [s][n])
  float* hb = lds + 65536;    // 8192: chain state 32x256
  const int tid   = threadIdx.x;
  const int wave  = tid >> 5, lane = tid & 31;
  const int lhalf = lane >> 4, lm = lane & 15;

  for (int i = tid; i < 65536 / 4; i += 512)
    ((float4*)M)[i] = ((const float4*)(ws + WS_ABT))[i];
  __syncthreads();

  // 7 squarings: wave w owns row-tile w (16 rows), all 16 col-tiles.
  for (int it = 0; it < 7; it++) {
    v8f accs[16];
    for (int nt = 0; nt < 16; nt++) {
      v8f acc = {};
#pragma unroll
      for (int kk = 0; kk < 64; kk++) {
        v2f a = *(const v2f*)(M + (wave * 16 + lm) * SD + kk * 4 + lhalf * 2);
        v2f b;
        b.x = M[(kk * 4 + lhalf * 2 + 0) * SD + nt * 16 + lm];
        b.y = M[(kk * 4 + lhalf * 2 + 1) * SD + nt * 16 + lm];
        acc = __builtin_amdgcn_wmma_f32_16x16x4_f32(false, a, false, b,
                                                    (short)0, acc, false, false);
      }
      accs[nt] = acc;
    }
    __syncthreads();                 // all reads of M done before overwrite
    for (int nt = 0; nt < 16; nt++) {
#pragma unroll
      for (int r = 0; r < 8; r++)
        M[(wave * 16 + r + lhalf * 8) * SD + nt * 16 + lm] = accs[nt][r];
    }
    __syncthreads();
  }

  // chunk chain
  for (int i = tid; i < 8192; i += 512) hb[i] = hidden[i];
  __syncthreads();
  const int mt = wave & 1;
  for (int c = 0; c < NCHUNK; c++) {
    for (int i = tid; i < 8192; i += 512) hstart[c * 8192 + i] = hb[i];
    v8f outq[2];
    for (int q = 0; q < 2; q++) {
      const int nt = (wave >> 1) + q * 8;
      v8f acc;
#pragma unroll
      for (int r = 0; r < 8; r++)
        acc[r] = sch[c * 8192 + (mt * 16 + r + lhalf * 8) * SD + nt * 16 + lm];
#pragma unroll
      for (int kk = 0; kk < 64; kk++) {
        v2f a = *(const v2f*)(hb + (mt * 16 + lm) * SD + kk * 4 + lhalf * 2);
        v2f b;
        b.x = M[(kk * 4 + lhalf * 2 + 0) * SD + nt * 16 + lm];
        b.y = M[(kk * 4 + lhalf * 2 + 1) * SD + nt * 16 + lm];
        acc = __builtin_amdgcn_wmma_f32_16x16x4_f32(false, a, false, b,
                                                    (short)0, acc, false, false);
      }
      outq[q] = acc;
    }
    __syncthreads();
    for (int q = 0; q < 2; q++) {
      const int nt = (wave >> 1) + q * 8;
#pragma unroll
      for (int r = 0; r < 8; r++)
        hb[(mt * 16 + r + lhalf * 8) * SD + nt * 16 + lm] = outq[q][r];
    }
    __syncthreads();
  }
  for (int i = tid; i < 8192; i += 512) hidden_final[i] = hb[i];
}

// ---------------------------------------------------------------------------
// K5: y = hs @ C^T + x @ D^T as flat (32*4096)x64 GEMM, K=256+64.
// 8 waves/WG; wave -> (row-tile, col-tile). B-operands broadcast from L2/WGP$.
// ---------------------------------------------------------------------------
__global__ __launch_bounds__(256) void ssm_outproj(
    const float* __restrict__ hs, const float* __restrict__ x,
    const float* __restrict__ ws, float* __restrict__ out)
{
  const int tid   = threadIdx.x;
  const int wave  = tid >> 5, lane = tid & 31;
  const int lhalf = lane >> 4, lm = lane & 15;
  const int rt = blockIdx.x * 2 + (wave & 1);
  const int nt = wave >> 1;
  const float* CSW = ws + WS_CSW;
  const float* DSW = ws + WS_DSW;
  const size_t row = (size_t)rt * 16 + lm;

  v8f acc = {};
#pragma unroll
  for (int kk = 0; kk < 64; kk++) {
    v2f a = *(const v2f*)(hs + row * SD + kk * 4 + lhalf * 2);
    v2f b = *(const v2f*)(CSW + ((kk * 2 + lhalf) * OD + nt * 16 + lm) * 2);
    acc = __builtin_amdgcn_wmma_f32_16x16x4_f32(false, a, false, b,
                                                (short)0, acc, false, false);
  }
#pragma unroll
  for (int kk = 0; kk < 16; kk++) {
    v2f a = *(const v2f*)(x + row * IND + kk * 4 + lhalf * 2);
    v2f b = *(const v2f*)(DSW + ((kk * 2 + lhalf) * OD + nt * 16 + lm) * 2);
    acc = __builtin_amdgcn_wmma_f32_16x16x4_f32(false, a, false, b,
                                                (short)0, acc, false, false);
  }
#pragma unroll
  for (int r = 0; r < 8; r++)
    out[((size_t)rt * 16 + r + lhalf * 8) * OD + nt * 16 + lm] = acc[r];
}

// ---------------------------------------------------------------------------
extern "C" void kernel_launch(void* const* d_in, const int* in_sizes, int n_in,
                              void* d_out, int out_size, void* d_ws, size_t ws_size,
                              hipStream_t stream)
{
  const float* x      = (const float*)d_in[0];
  const float* hidden = (const float*)d_in[1];
  const float* A      = (const float*)d_in[2];
  const float* B      = (const float*)d_in[3];
  const float* C      = (const float*)d_in[4];
  const float* D      = (const float*)d_in[5];
  const float* log_dt = (const float*)d_in[6];
  float* out = (float*)d_out;
  float* ws  = (float*)d_ws;     // needs ~137 MB (hs dominates)

  (void)hipFuncSetAttribute((const void*)ssm_scan,
                            hipFuncAttributeMaxDynamicSharedMemorySize, LDS_BYTES);
  (void)hipFuncSetAttribute((const void*)ssm_combine,
                            hipFuncAttributeMaxDynamicSharedMemorySize, LDS_BYTES);

  ssm_setup<<<(167936 + 255) / 256, 256, 0, stream>>>(A, B, C, D, log_dt, ws);
  // pass A: 64 parallel local chunk scans (h0 = 0) -> S_c
  ssm_scan<<<NCHUNK * 2, 512, LDS_BYTES, stream>>>(x, ws, ws + WS_SCH, ws + WS_HST,
                                                   ws + WS_HS, 0);
  // pass B: chunk chain -> Hstart[], hidden_final (d_out tail)
  ssm_combine<<<1, 512, LDS_BYTES, stream>>>(ws, hidden, ws + WS_HST, ws + WS_SCH,
                                             out + (size_t)BATCH * SEQT * OD);
  // pass C: corrected chunk scans, stream h_t -> hs
  ssm_scan<<<NCHUNK * 2, 512, LDS_BYTES, stream>>>(x, ws, ws + WS_SCH, ws + WS_HST,
                                                   ws + WS_HS, 1);
  // output projection
  ssm_outproj<<<(BATCH * SEQT) / 32, 256, 0, stream>>>(ws + WS_HS, x, ws, out);
}